// LightGCN_73821897883700
// MI455X (gfx1250) — compile-verified
//
#include <hip/hip_runtime.h>
#include <hip/hip_bf16.h>

#define N_USERS 100000
#define N_ITEMS 50000
#define N_NODES (N_USERS + N_ITEMS)
#define EMB_DIM 64
#define N_EDGES 4000000
#define N_LAYERS 3
#define BATCH 4096

typedef __attribute__((ext_vector_type(2))) float v2f;
typedef __attribute__((ext_vector_type(8))) float v8f;

// ---------------------------------------------------------------------------
// Kernel 1: cur = acc = concat(user_emb, item_emb); nxt = 0.  float4 vectorized.
// ---------------------------------------------------------------------------
__global__ void lgcn_init(const float4* __restrict__ ue, const float4* __restrict__ ie,
                          float4* __restrict__ cur, float4* __restrict__ acc,
                          float4* __restrict__ nxt) {
  const int n_user4 = N_USERS * (EMB_DIM / 4);
  const int total4  = N_NODES * (EMB_DIM / 4);
  int i = blockIdx.x * blockDim.x + threadIdx.x;
  if (i >= total4) return;
  float4 v = (i < n_user4) ? ue[i] : ie[i - n_user4];
  cur[i] = v;
  acc[i] = v;
  nxt[i] = make_float4(0.f, 0.f, 0.f, 0.f);
}

// ---------------------------------------------------------------------------
// Kernel 2: COO SpMM scatter: nxt[dst] += val * cur[src].
// 16 lanes cooperate per edge; each lane moves one float4 of the 64-float row
// and issues 4 no-return global_atomic_add_f32 into L2 (tables are L2-resident).
// ---------------------------------------------------------------------------
__global__ void lgcn_spmm_scatter(const float* __restrict__ vals,
                                  const int* __restrict__ src,
                                  const int* __restrict__ dst,
                                  const float* __restrict__ cur,
                                  float* __restrict__ nxt) {
  int t    = blockIdx.x * blockDim.x + threadIdx.x;
  int e    = t >> 4;        // edge id
  int lane = t & 15;        // float4 slot within the row
  if (e >= N_EDGES) return;
  float v = vals[e];
  int   s = src[e];
  int   d = dst[e];
  float4 m = ((const float4*)(cur + (size_t)s * EMB_DIM))[lane];
  float* drow = nxt + (size_t)d * EMB_DIM + lane * 4;
  atomicAdd(drow + 0, v * m.x);
  atomicAdd(drow + 1, v * m.y);
  atomicAdd(drow + 2, v * m.z);
  atomicAdd(drow + 3, v * m.w);
}

// ---------------------------------------------------------------------------
// Kernel 3: acc += nxt; zero old cur (it becomes next layer's scatter target).
// Host swaps cur/nxt pointers after this kernel.
// ---------------------------------------------------------------------------
__global__ void lgcn_acc_add_zero(float4* __restrict__ acc,
                                  const float4* __restrict__ nxt,
                                  float4* __restrict__ oldcur) {
  const int total4 = N_NODES * (EMB_DIM / 4);
  int i = blockIdx.x * blockDim.x + threadIdx.x;
  if (i >= total4) return;
  float4 a = acc[i];
  float4 n = nxt[i];
  acc[i]    = make_float4(a.x + n.x, a.y + n.y, a.z + n.z, a.w + n.w);
  oldcur[i] = make_float4(0.f, 0.f, 0.f, 0.f);
}

// ---------------------------------------------------------------------------
// Kernel 4: batched dot products via V_WMMA_F32_16X16X4_F32.
// One wave32 handles 16 batch pairs: A[m,k] = accU[users[b0+m], k],
// B[k,n] = accI[items[b0+n], k].  f32 operand striping (ISA 7.12.2):
//   A/B: lanes 0-15 hold K=kk+0 (vgpr0) / kk+1 (vgpr1); lanes 16-31 hold kk+2/kk+3.
// Accumulate over K=64 (16 WMMAs), D tile -> LDS, read diagonal, scale by
// 1/(N_LAYERS+1)^2 = 1/16 (both operands carry the un-divided layer sum).
// ---------------------------------------------------------------------------
__global__ void lgcn_dot_wmma(const float* __restrict__ acc,
                              const int* __restrict__ users,
                              const int* __restrict__ items,
                              float* __restrict__ out) {
  __shared__ float tile[4][16][17];   // [wave][M][N] with pad
  int wave = threadIdx.x >> 5;        // 0..3
  int lane = threadIdx.x & 31;
  int t16  = blockIdx.x * 4 + wave;   // tile id, 0..255 (BATCH/16 tiles)
  int m    = lane & 15;
  int half = lane >> 4;               // 0: K slots 0,1   1: K slots 2,3
  int b    = t16 * 16 + m;

  int ub = users[b];
  int ib = items[b];
  const float* urow = acc + (size_t)ub * EMB_DIM;
  const float* irow = acc + ((size_t)N_USERS + (size_t)ib) * EMB_DIM;

  v8f c = {};
  #pragma unroll
  for (int kk = 0; kk < EMB_DIM; kk += 4) {
    int k = kk + half * 2;
    v2f a;  a[0]  = urow[k]; a[1]  = urow[k + 1];
    v2f bb; bb[0] = irow[k]; bb[1] = irow[k + 1];
    c = __builtin_amdgcn_wmma_f32_16x16x4_f32(
        /*neg_a=*/false, a, /*neg_b=*/false, bb,
        /*c_mod=*/(short)0, c, /*reuse_a=*/false, /*reuse_b=*/false);
  }

  // D layout: VGPR j holds M=j (lanes 0-15) or M=j+8 (lanes 16-31), N=lane%16.
  #pragma unroll
  for (int j = 0; j < 8; ++j)
    tile[wave][j + half * 8][m] = c[j];
  __syncthreads();

  if (lane < 16) {
    const float scale = 1.0f / ((N_LAYERS + 1) * (N_LAYERS + 1)); // 1/16
    out[b] = tile[wave][m][m] * scale;
  }
}

// ---------------------------------------------------------------------------
extern "C" void kernel_launch(void* const* d_in, const int* in_sizes, int n_in,
                              void* d_out, int out_size, void* d_ws, size_t ws_size,
                              hipStream_t stream) {
  const float* user_emb = (const float*)d_in[0];
  const float* item_emb = (const float*)d_in[1];
  const float* vals     = (const float*)d_in[2];
  const int*   src      = (const int*)d_in[3];
  const int*   dst      = (const int*)d_in[4];
  const int*   users    = (const int*)d_in[5];
  const int*   items    = (const int*)d_in[6];
  float*       out      = (float*)d_out;

  const size_t nfloats = (size_t)N_NODES * EMB_DIM;   // 9.6M floats (38.4 MB)
  float* cur = (float*)d_ws;
  float* nxt = cur + nfloats;
  float* acc = nxt + nfloats;                         // total ws use: 115.2 MB

  const int total4 = N_NODES * (EMB_DIM / 4);         // 2.4M float4s
  const int tpb = 256;

  lgcn_init<<<(total4 + tpb - 1) / tpb, tpb, 0, stream>>>(
      (const float4*)user_emb, (const float4*)item_emb,
      (float4*)cur, (float4*)acc, (float4*)nxt);

  for (int layer = 0; layer < N_LAYERS; ++layer) {
    const long long nthreads = (long long)N_EDGES * 16;           // 64M
    lgcn_spmm_scatter<<<(int)(nthreads / tpb), tpb, 0, stream>>>(
        vals, src, dst, cur, nxt);
    lgcn_acc_add_zero<<<(total4 + tpb - 1) / tpb, tpb, 0, stream>>>(
        (float4*)acc, (const float4*)nxt, (float4*)cur);
    float* tmp = cur; cur = nxt; nxt = tmp;           // deterministic pointer swap
  }

  // BATCH/16 = 256 tiles, 4 wave32 tiles per 128-thread block -> 64 blocks.
  lgcn_dot_wmma<<<BATCH / 64, 128, 0, stream>>>(acc, users, items, out);
}